// TemporalCrossTransformer_41832981463590
// MI455X (gfx1250) — compile-verified
//
#include <hip/hip_runtime.h>
#include <hip/hip_bf16.h>
#include <math.h>

// ---------------------------------------------------------------------------
// TemporalCrossTransformer for MI455X (gfx1250, wave32, WMMA)
//
// Sizes: SEQ=16, D_IN=2048, TSS=3, D_OUT=1128, T=C(16,3)=560, C=5, Q=1
//
// Factorization: s_t @ W.T  ==  sum_j P_j[tup[:,j]],  P_j = x @ W_j.T
// so the projection GEMM is only M=16 rows per batch (35x FLOP reduction),
// and is bound by the 55MB weight read (~2.4us @ 23.3 TB/s HBM).
// All GEMMs use V_WMMA_F32_16X16X4_F32 (fp32 in / fp32 accumulate).
//
// Each wave computes TWO 16-wide output tiles per A load (1.5 loads/WMMA).
// N-tails: clamp source rows (branch-free K-loops, EXEC all-1s), guard only
// stores/accumulations.
// ---------------------------------------------------------------------------

#define SEQ     16
#define D_IN    2048
#define TSS     3
#define D_OUT   1128
#define TUP     560
#define NSUP    5
#define NBATCH  6          // 5 support + 1 query
#define PCOLS   (TSS * D_OUT)        // 3384 = packed P_j columns
#define NTILE_O ((D_OUT + 15) / 16)  // 71 (last tile has 8 valid cols)
#define NPAIR_O ((NTILE_O + 1) / 2)  // 36 column-tile pairs
#define TQTILES (TUP / 16)           // 35
#define NPAIR_T ((TQTILES + 1) / 2)  // 18 ts-tile pairs
#define LN_EPS  1e-5f

typedef float v2f __attribute__((ext_vector_type(2)));
typedef float v8f __attribute__((ext_vector_type(8)));

// D(16x16,f32) = A(16x4,f32) * B(4x16,f32) + C
// Lane mapping (wave32): l15 = lane&15, kb = 2*(lane>>4)
//   A: a.x = A[m=l15][kb],  a.y = A[m=l15][kb+1]
//   B: b.x = B[kb][n=l15],  b.y = B[kb+1][n=l15]
//   C/D: c[i] = C[m = i + 8*(lane>>4)][n=l15]
__device__ __forceinline__ v8f wmma_f32(v2f a, v2f b, v8f c) {
  return __builtin_amdgcn_wmma_f32_16x16x4_f32(
      /*neg_a=*/false, a, /*neg_b=*/false, b,
      /*c_mod=*/(short)0, c, /*reuse_a=*/false, /*reuse_b=*/false);
}

__device__ __forceinline__ v2f ld2(const float* p) {
  v2f r;
  r.x = p[0];
  r.y = p[1];
  return r;
}

__device__ __forceinline__ int imin(int a, int b) { return a < b ? a : b; }

// lexicographic unranking of combinations(16,3)
__device__ __forceinline__ void unrank3(int t, int& a, int& b, int& c) {
  int idx = t;
  a = 0;
  for (; a < SEQ; ++a) {
    int cnt = (15 - a) * (14 - a) / 2;   // C(15-a,2)
    if (idx < cnt) break;
    idx -= cnt;
  }
  b = a + 1;
  for (; b < SEQ; ++b) {
    int cnt = 15 - b;                    // C(15-b,1)
    if (idx < cnt) break;
    idx -= cnt;
  }
  c = b + 1 + idx;
}

// ---------------------------------------------------------------------------
// K1: x[n] = input[n] + positional_encoding ; also zero distance accumulators
// grid: (NBATCH, SEQ) x 256
// ---------------------------------------------------------------------------
__global__ void tct_build_x(const float* __restrict__ sup,
                            const float* __restrict__ qry,
                            float* __restrict__ X,
                            float* __restrict__ DIST) {
  const int n = blockIdx.x;       // 0..5
  const int s = blockIdx.y;       // 0..15
  const float* src = (n < NSUP) ? (sup + ((size_t)n * SEQ + s) * D_IN)
                                : (qry + (size_t)s * D_IN);
  float* dst = X + ((size_t)n * SEQ + s) * D_IN;
  const float negk = -logf(10000.0f) / (float)D_IN;
  for (int d = threadIdx.x; d < D_IN; d += blockDim.x) {
    int   i   = d >> 1;
    float div = expf((float)(2 * i) * negk);
    float ang = (float)s * div;
    float pe  = ((d & 1) ? cosf(ang) : sinf(ang)) * 0.1f;
    dst[d] = src[d] + pe;
  }
  if (n == 0 && s == 0 && threadIdx.x < NSUP) DIST[threadIdx.x] = 0.0f;
}

// ---------------------------------------------------------------------------
// K2: projection GEMM  P[n][s][j*1128+o] = sum_k X[n][s][k] * W[o][j*2048+k]
// One wave per PAIR of 16x16 output tiles (A load amortized over 2 WMMAs).
// grid: (ceil(36/4)=9, j=3, n*2+kv=12) x 128 (4 waves)
// ---------------------------------------------------------------------------
__global__ __launch_bounds__(128)
void tct_proj(const float* __restrict__ X,
              const float* __restrict__ wk,
              const float* __restrict__ wv,
              float* __restrict__ PK,
              float* __restrict__ PV) {
  const int wave = threadIdx.x >> 5;
  const int lane = threadIdx.x & 31;
  const int pair = blockIdx.x * 4 + wave;     // 0..35
  if (pair >= NPAIR_O) return;                // whole-wave exit, no barriers used
  const int j  = blockIdx.y;                  // 0..2
  const int n  = blockIdx.z >> 1;             // 0..5
  const int kv = blockIdx.z & 1;              // 0:k 1:v
  const float* W = kv ? wv : wk;
  float*       P = kv ? PV : PK;

  const int l15 = lane & 15;
  const int kb  = (lane >> 4) << 1;           // 0 or 2
  const int hi  = (lane >> 4) << 3;           // 0 or 8
  const int o0  = pair * 32 + l15;
  const int o1  = o0 + 16;
  const bool v0 = (o0 < D_OUT);
  const bool v1 = (o1 < D_OUT);

  const float* Arow = X + ((size_t)n * SEQ + l15) * D_IN + kb;          // A row m
  const float* B0 = W + (size_t)imin(o0, D_OUT - 1) * (TSS * D_IN) + (size_t)j * D_IN + kb;
  const float* B1 = W + (size_t)imin(o1, D_OUT - 1) * (TSS * D_IN) + (size_t)j * D_IN + kb;

  v8f c0 = {}, c1 = {};
  #pragma unroll 8
  for (int k = 0; k < D_IN; k += 4) {
    v2f a = ld2(Arow + k);
    c0 = wmma_f32(a, ld2(B0 + k), c0);
    c1 = wmma_f32(a, ld2(B1 + k), c1);
  }

  float* Pb = P + (size_t)n * SEQ * PCOLS + (size_t)j * D_OUT;
  if (v0) {
    #pragma unroll
    for (int i = 0; i < 8; ++i)
      Pb[(size_t)(i + hi) * PCOLS + o0] = c0[i];
  }
  if (v1) {
    #pragma unroll
    for (int i = 0; i < 8; ++i)
      Pb[(size_t)(i + hi) * PCOLS + o1] = c1[i];
  }
}

// ---------------------------------------------------------------------------
// K3: tuple combine + bias, LayerNorm(k), write KS/VS [n][t][D_OUT]
// grid: (560, 6) x 256
// ---------------------------------------------------------------------------
__global__ __launch_bounds__(256)
void tct_combine(const float* __restrict__ PK, const float* __restrict__ PV,
                 const float* __restrict__ bk, const float* __restrict__ bv,
                 const float* __restrict__ gamma, const float* __restrict__ beta,
                 float* __restrict__ KS, float* __restrict__ VS) {
  const int t = blockIdx.x;    // tuple
  const int n = blockIdx.y;    // batch
  int ia, ib, ic;
  unrank3(t, ia, ib, ic);

  __shared__ float kbuf[D_OUT];
  __shared__ float red[256];
  __shared__ float red2[256];

  const float* Pk = PK + (size_t)n * SEQ * PCOLS;
  const float* Pv = PV + (size_t)n * SEQ * PCOLS;
  float lsum = 0.0f, lsq = 0.0f;
  for (int o = threadIdx.x; o < D_OUT; o += blockDim.x) {
    float kp = Pk[(size_t)ia * PCOLS + o]
             + Pk[(size_t)ib * PCOLS + D_OUT + o]
             + Pk[(size_t)ic * PCOLS + 2 * D_OUT + o] + bk[o];
    kbuf[o] = kp;
    lsum += kp;
    lsq  += kp * kp;
    float vp = Pv[(size_t)ia * PCOLS + o]
             + Pv[(size_t)ib * PCOLS + D_OUT + o]
             + Pv[(size_t)ic * PCOLS + 2 * D_OUT + o] + bv[o];
    VS[((size_t)n * TUP + t) * D_OUT + o] = vp;
  }
  red[threadIdx.x]  = lsum;
  red2[threadIdx.x] = lsq;
  __syncthreads();
  for (int s = 128; s > 0; s >>= 1) {
    if (threadIdx.x < s) {
      red[threadIdx.x]  += red[threadIdx.x + s];
      red2[threadIdx.x] += red2[threadIdx.x + s];
    }
    __syncthreads();
  }
  const float mu  = red[0] / (float)D_OUT;
  const float var = red2[0] / (float)D_OUT - mu * mu;
  const float inv = rsqrtf(var + LN_EPS);
  for (int o = threadIdx.x; o < D_OUT; o += blockDim.x)
    KS[((size_t)n * TUP + t) * D_OUT + o] = (kbuf[o] - mu) * inv * gamma[o] + beta[o];
}

// ---------------------------------------------------------------------------
// K4: per (16-query-tile, class) flash-style attention + squared distance.
//   phase 1: scores[16][560] = (q_ks_tile @ ck^T)/sqrt(D) via WMMA -> LDS
//   phase 2: row softmax in LDS
//   phase 3: proto = attn @ cv via WMMA (A from LDS), fused ||q_vs-proto||^2
// grid: (35, 5) x 256 (8 waves); two tiles per wave per A load.
// ---------------------------------------------------------------------------
#define SC_STRIDE 564   // 560 + pad (stride%64banks -> conflict-free col reads)

__global__ __launch_bounds__(256)
void tct_attn(const float* __restrict__ KS, const float* __restrict__ VS,
              const int* __restrict__ labels, float* __restrict__ DIST) {
  const int tq0 = blockIdx.x * 16;
  const int cls = blockIdx.y;
  const int lab = labels[cls];

  const float* QK = KS + (size_t)NSUP * TUP * D_OUT;
  const float* QV = VS + (size_t)NSUP * TUP * D_OUT;
  const float* CK = KS + (size_t)lab * TUP * D_OUT;
  const float* CV = VS + (size_t)lab * TUP * D_OUT;

  __shared__ float sc[16][SC_STRIDE];
  __shared__ float red[16][17];
  __shared__ float rstat[16];
  __shared__ float bred[256];

  const int wave = threadIdx.x >> 5;
  const int lane = threadIdx.x & 31;
  const int l15  = lane & 15;
  const int kb   = (lane >> 4) << 1;
  const int hi   = (lane >> 4) << 3;
  const float scale = rsqrtf((float)D_OUT);

  // -------- phase 1: scores strip [16 x 560], 2 ts-tiles per A load --------
  const float* Aq = QK + (size_t)(tq0 + l15) * D_OUT + kb;   // A row m = query tuple
  for (int pp = wave; pp < NPAIR_T; pp += 8) {
    const int ts0 = pp * 32;
    const int r0  = ts0 + l15;                  // always < 560
    const int r1  = ts0 + 16 + l15;             // invalid for pp==17
    const float* B0 = CK + (size_t)r0 * D_OUT + kb;
    const float* B1 = CK + (size_t)imin(r1, TUP - 1) * D_OUT + kb;
    v8f c0 = {}, c1 = {};
    #pragma unroll 8
    for (int k = 0; k < D_OUT; k += 4) {
      v2f a = ld2(Aq + k);
      c0 = wmma_f32(a, ld2(B0 + k), c0);
      c1 = wmma_f32(a, ld2(B1 + k), c1);
    }
    #pragma unroll
    for (int i = 0; i < 8; ++i)
      sc[i + hi][ts0 + l15] = c0[i] * scale;
    if (ts0 + 16 < TUP) {
      #pragma unroll
      for (int i = 0; i < 8; ++i)
        sc[i + hi][ts0 + 16 + l15] = c1[i] * scale;
    }
  }
  __syncthreads();

  // -------- phase 2: softmax over support axis (560) per row --------
  {
    const int row = threadIdx.x >> 4;   // 16 rows x 16 threads
    const int col = threadIdx.x & 15;
    float m = -3.402823466e38f;
    for (int ts = col; ts < TUP; ts += 16) m = fmaxf(m, sc[row][ts]);
    red[row][col] = m;
    __syncthreads();
    if (col == 0) {
      float mm = red[row][0];
      #pragma unroll
      for (int i = 1; i < 16; ++i) mm = fmaxf(mm, red[row][i]);
      rstat[row] = mm;
    }
    __syncthreads();
    const float mm = rstat[row];
    float s = 0.0f;
    for (int ts = col; ts < TUP; ts += 16) {
      float e = __expf(sc[row][ts] - mm);
      sc[row][ts] = e;
      s += e;
    }
    red[row][col] = s;
    __syncthreads();
    if (col == 0) {
      float ss = 0.0f;
      #pragma unroll
      for (int i = 0; i < 16; ++i) ss += red[row][i];
      rstat[row] = 1.0f / ss;
    }
    __syncthreads();
    const float is = rstat[row];
    for (int ts = col; ts < TUP; ts += 16) sc[row][ts] *= is;
  }
  __syncthreads();

  // -------- phase 3: proto = attn @ cv, 2 col-tiles per LDS A load --------
  float acc = 0.0f;
  const float* As = &sc[l15][kb];
  for (int cp = wave; cp < NPAIR_O; cp += 8) {
    const int  o0 = cp * 32 + l15;
    const int  o1 = o0 + 16;
    const bool v0 = (o0 < D_OUT);
    const bool v1 = (o1 < D_OUT);
    const float* B0 = CV + (size_t)kb * D_OUT + imin(o0, D_OUT - 1);
    const float* B1 = CV + (size_t)kb * D_OUT + imin(o1, D_OUT - 1);
    v8f c0 = {}, c1 = {};
    #pragma unroll 8
    for (int k = 0; k < TUP; k += 4) {
      v2f a, b0, b1;
      a.x = As[k];                       // A[m][k] = attn weights from LDS
      a.y = As[k + 1];
      b0.x = B0[(size_t)k * D_OUT];
      b0.y = B0[(size_t)k * D_OUT + D_OUT];
      b1.x = B1[(size_t)k * D_OUT];
      b1.y = B1[(size_t)k * D_OUT + D_OUT];
      c0 = wmma_f32(a, b0, c0);
      c1 = wmma_f32(a, b1, c1);
    }
    if (v0) {
      #pragma unroll
      for (int i = 0; i < 8; ++i) {
        float d = QV[(size_t)(tq0 + i + hi) * D_OUT + o0] - c0[i];
        acc += d * d;
      }
    }
    if (v1) {
      #pragma unroll
      for (int i = 0; i < 8; ++i) {
        float d = QV[(size_t)(tq0 + i + hi) * D_OUT + o1] - c1[i];
        acc += d * d;
      }
    }
  }
  bred[threadIdx.x] = acc;
  __syncthreads();
  for (int s = 128; s > 0; s >>= 1) {
    if (threadIdx.x < s) bred[threadIdx.x] += bred[threadIdx.x + s];
    __syncthreads();
  }
  if (threadIdx.x == 0) atomicAdd(&DIST[cls], bred[0]);
}

// ---------------------------------------------------------------------------
// K5: logits[q=0][c] = -dist[c] / T
// ---------------------------------------------------------------------------
__global__ void tct_finalize(const float* __restrict__ DIST, float* __restrict__ out) {
  if (threadIdx.x < NSUP) out[threadIdx.x] = -DIST[threadIdx.x] / (float)TUP;
}

// ---------------------------------------------------------------------------
extern "C" void kernel_launch(void* const* d_in, const int* in_sizes, int n_in,
                              void* d_out, int out_size, void* d_ws, size_t ws_size,
                              hipStream_t stream) {
  const float* sup    = (const float*)d_in[0];
  const float* qry    = (const float*)d_in[1];
  const int*   labels = (const int*)  d_in[2];
  const float* wk     = (const float*)d_in[3];
  const float* bk     = (const float*)d_in[4];
  const float* wv     = (const float*)d_in[5];
  const float* bv     = (const float*)d_in[6];
  const float* gamma  = (const float*)d_in[7];
  const float* beta   = (const float*)d_in[8];

  float* ws = (float*)d_ws;
  float* X    = ws;                                   // 6*16*2048
  float* PK   = X  + (size_t)NBATCH * SEQ * D_IN;     // 6*16*3384
  float* PV   = PK + (size_t)NBATCH * SEQ * PCOLS;    // 6*16*3384
  float* KS   = PV + (size_t)NBATCH * SEQ * PCOLS;    // 6*560*1128
  float* VS   = KS + (size_t)NBATCH * TUP * D_OUT;    // 6*560*1128
  float* DIST = VS + (size_t)NBATCH * TUP * D_OUT;    // 5

  tct_build_x <<<dim3(NBATCH, SEQ),            256, 0, stream>>>(sup, qry, X, DIST);
  tct_proj    <<<dim3((NPAIR_O + 3) / 4, TSS, NBATCH * 2), 128, 0, stream>>>(X, wk, wv, PK, PV);
  tct_combine <<<dim3(TUP, NBATCH),            256, 0, stream>>>(PK, PV, bk, bv, gamma, beta, KS, VS);
  tct_attn    <<<dim3(TQTILES, NSUP),          256, 0, stream>>>(KS, VS, labels, DIST);
  tct_finalize<<<1, 32, 0, stream>>>(DIST, (float*)d_out);
}